// BaseTextureDiffusion_24592982736955
// MI455X (gfx1250) — compile-verified
//
#include <hip/hip_runtime.h>
#include <stdint.h>

// Problem constants (fixed by the reference).
#define Bc   2
#define Cc   24
#define Hc   256
#define Wc   256
#define Rr   7
#define PADc 3
#define HWc  (Hc * Wc)

// Tiling.
#define TILE_W   64          // output tile width  (16 threads x 4 px)
#define TILE_H   16          // output tile height
#define LDS_W    (TILE_W + Rr - 1)   // 70 staged columns
#define LDS_H    (TILE_H + Rr - 1)   // 22 staged rows
// Row stride 74: 296B rows -> every row 8B-aligned (enables ds_load_b64
// merging), and 74 mod 64 = 10 banks puts the two 16-lane wave halves in
// different (mod 4) bank residue classes -> conflict-free sliding reads.
#define LDS_STRIDE 74
#define NTHREADS 256

typedef float v4f __attribute__((ext_vector_type(4)));

__global__ __launch_bounds__(NTHREADS)
void texdiff_kernel(const float* __restrict__ latent,
                    const float* __restrict__ weights,
                    float* __restrict__ out)
{
    __shared__ __align__(16) float tile[LDS_H * LDS_STRIDE];

    const int plane = blockIdx.z;                // b*C + c
    const int tileX = blockIdx.x * TILE_W;
    const int tileY = blockIdx.y * TILE_H;
    const int tid   = threadIdx.x;

    const float* lat_plane = latent + (size_t)plane * HWc;

    // ---- Stage replicate-padded latent tile into LDS via CDNA5 async loads
    // (GLOBAL_LOAD_ASYNC_TO_LDS_B32, tracked by ASYNCcnt). Latent is 12 MB
    // total -> L2-resident; async path bypasses VGPRs entirely.
    for (int idx = tid; idx < LDS_H * LDS_W; idx += NTHREADS) {
        int gy = idx / LDS_W;
        int gx = idx - gy * LDS_W;
        int gh = tileY + gy - PADc; gh = min(max(gh, 0), Hc - 1);
        int gw = tileX + gx - PADc; gw = min(max(gw, 0), Wc - 1);
        // Low 32 bits of a generic pointer to __shared__ are the LDS offset.
        unsigned lds_off = (unsigned)(uint64_t)(uintptr_t)(&tile[gy * LDS_STRIDE + gx]);
        uint64_t gaddr   = (uint64_t)(uintptr_t)(lat_plane + gh * Wc + gw);
        asm volatile("global_load_async_to_lds_b32 %0, %1, off"
                     :: "v"(lds_off), "v"(gaddr)
                     : "memory");
    }
    // Wait for this wave's async LDS writes; the barrier then makes all
    // waves' staged data visible workgroup-wide.
    asm volatile("s_wait_asynccnt 0x0" ::: "memory");
    __syncthreads();

    // ---- Compute: each thread produces 4 consecutive output pixels.
    const int tx = tid & 15;       // 0..15  (x group of 4 px)
    const int ty = tid >> 4;       // 0..15  (row)
    const int w0 = tileX + tx * 4;
    const int h  = tileY + ty;
    const int lx = tx * 4;

    // weights[b,c,k,h,w] flat: plane*49*HW + k*HW + h*W + w  (16B-aligned at w0)
    const v4f* wbase = (const v4f*)(weights + (size_t)plane * (Rr * Rr) * HWc
                                            + (size_t)h * Wc + w0);
    const size_t kstride4 = (size_t)HWc / 4;

    v4f acc = (v4f)(0.0f);
    #pragma unroll
    for (int i = 0; i < Rr; ++i) {
        // 10-float sliding window for this kernel row, loaded once from LDS
        // (8B-aligned base -> compiler can emit ds_load_b64 pairs).
        float r[Rr - 1 + 4];
        const float* trow = &tile[(ty + i) * LDS_STRIDE + lx];
        #pragma unroll
        for (int t = 0; t < Rr - 1 + 4; ++t) r[t] = trow[t];

        #pragma unroll
        for (int j = 0; j < Rr; ++j) {
            // 588 MB single-use stream: non-temporal so it doesn't evict the
            // L2-resident latent planes.
            v4f wk = __builtin_nontemporal_load(&wbase[(size_t)(i * Rr + j) * kstride4]);
            acc.x += wk.x * r[j + 0];
            acc.y += wk.y * r[j + 1];
            acc.z += wk.z * r[j + 2];
            acc.w += wk.w * r[j + 3];
        }
    }

    // Output is write-once (12 MB): NT store, don't pollute L2.
    v4f* optr = (v4f*)(out + (size_t)plane * HWc + (size_t)h * Wc + w0);
    __builtin_nontemporal_store(acc, optr);
}

extern "C" void kernel_launch(void* const* d_in, const int* in_sizes, int n_in,
                              void* d_out, int out_size, void* d_ws, size_t ws_size,
                              hipStream_t stream) {
    (void)in_sizes; (void)n_in; (void)d_ws; (void)ws_size; (void)out_size;
    const float* latent  = (const float*)d_in[0];
    const float* weights = (const float*)d_in[1];
    // d_in[2] = window_size (==7), baked in as compile-time constant.
    float* out = (float*)d_out;

    dim3 grid(Wc / TILE_W, Hc / TILE_H, Bc * Cc);   // 4 x 16 x 48
    dim3 block(NTHREADS);
    texdiff_kernel<<<grid, block, 0, stream>>>(latent, weights, out);
}